// Contrast_loss_point_cloud_inetra_batch_70282844832077
// MI455X (gfx1250) — compile-verified
//
#include <hip/hip_runtime.h>
#include <math.h>

typedef __attribute__((ext_vector_type(2))) float v2f;
typedef __attribute__((ext_vector_type(8))) float v8f;

#define M_TOTAL 8192
#define C_DIM 64
#define NUM_TILES 512   // M_TOTAL / 16

// exp(sim / 0.07) == exp2(sim * log2(e)/0.07); constant folded in double.
#define EXP2_SCALE ((float)(1.4426950408889634 / 0.07))

// ---------------- JAX threefry2x32, key = jax.random.key(42) -> {0, 42} ----
__device__ __forceinline__ unsigned rotl32(unsigned v, int r) {
  return (v << r) | (v >> (32 - r));
}

__device__ __forceinline__ void threefry2x32_42(unsigned c0, unsigned c1,
                                                unsigned &o0, unsigned &o1) {
  const unsigned ks0 = 0u, ks1 = 42u;
  const unsigned ks2 = ks0 ^ ks1 ^ 0x1BD11BDAu;
  unsigned x0 = c0 + ks0;
  unsigned x1 = c1 + ks1;
#define TF_R(r) { x0 += x1; x1 = rotl32(x1, r); x1 ^= x0; }
  TF_R(13) TF_R(15) TF_R(26) TF_R(6)
  x0 += ks1; x1 += ks2 + 1u;
  TF_R(17) TF_R(29) TF_R(16) TF_R(24)
  x0 += ks2; x1 += ks0 + 2u;
  TF_R(13) TF_R(15) TF_R(26) TF_R(6)
  x0 += ks0; x1 += ks1 + 3u;
  TF_R(17) TF_R(29) TF_R(16) TF_R(24)
  x0 += ks1; x1 += ks2 + 4u;
  TF_R(13) TF_R(15) TF_R(26) TF_R(6)
  x0 += ks2; x1 += ks0 + 5u;
#undef TF_R
  o0 = x0; o1 = x1;
}

// ---------------- Kernel 1: bincount -> keep_p -> threefry bernoulli -> meta
// meta[m] = label(2b) | (sel << 2); also writes n_selected.
__global__ __launch_bounds__(1024)
void setup_meta_kernel(const int* __restrict__ labels,
                       int* __restrict__ meta,
                       int* __restrict__ nsel) {
  __shared__ int cnt[4];
  __shared__ int red[1024];
  const int tid = threadIdx.x;
  if (tid < 4) cnt[tid] = 0;
  __syncthreads();
  for (int i = tid; i < M_TOTAL; i += 1024)
    atomicAdd(&cnt[labels[i] & 3], 1);
  __syncthreads();
  float keep[4];
#pragma unroll
  for (int c = 0; c < 4; c++) {
    float p = 750.0f / ((float)cnt[c] + 1.0f);
    keep[c] = p < 1.0f ? p : 1.0f;
  }
  int local = 0;
#pragma unroll
  for (int q = 0; q < 4; q++) {
    const int i = tid + q * 1024;          // counter index in [0, 4096)
    unsigned b0, b1;
    threefry2x32_42((unsigned)i, (unsigned)(i + 4096), b0, b1);
    const float u0 = __uint_as_float((b0 >> 9) | 0x3f800000u) - 1.0f;
    const float u1 = __uint_as_float((b1 >> 9) | 0x3f800000u) - 1.0f;
    const int l0 = labels[i] & 3;
    const int l1 = labels[i + 4096] & 3;
    const int s0 = (u0 < keep[l0]) ? 1 : 0;
    const int s1 = (u1 < keep[l1]) ? 1 : 0;
    meta[i]        = l0 | (s0 << 2);
    meta[i + 4096] = l1 | (s1 << 2);
    local += s0 + s1;
  }
  red[tid] = local;
  __syncthreads();
  for (int off = 512; off > 0; off >>= 1) {
    if (tid < off) red[tid] += red[tid + off];
    __syncthreads();
  }
  if (tid == 0) nsel[0] = red[0];
}

// ---------------- Kernel 2: transpose + L2 normalize
// feats[m][c] lives at input[c*8192 + m]; produce row-major nv[m*64 + c].
__global__ __launch_bounds__(256)
void normalize_kernel(const float* __restrict__ feats, float* __restrict__ nv) {
  const int m = blockIdx.x * blockDim.x + threadIdx.x;
  float f[C_DIM];
  float ss = 0.0f;
#pragma unroll
  for (int c = 0; c < C_DIM; c++) {
    const float v = feats[c * M_TOTAL + m];
    f[c] = v;
    ss += v * v;
  }
  const float nrm = sqrtf(ss);
#pragma unroll
  for (int c = 0; c < C_DIM; c++)
    nv[m * C_DIM + c] = f[c] / nrm;
}

// ---------------- Kernel 3: fused WMMA GEMM + exp + mask + row reduction
// Wave w owns 16 output rows; sweeps all 512 column tiles.
__global__ __launch_bounds__(256)
void contrast_main_kernel(const float* __restrict__ nv,
                          const int* __restrict__ meta,
                          float* __restrict__ per) {
  const int lane = threadIdx.x & 31;
  const int wave = threadIdx.x >> 5;
  const int tileI = blockIdx.x * 8 + wave;      // 0 .. 511
  const int rowBase = tileI * 16;
  const int lo = lane & 15;
  const int hi = lane >> 4;

  // A slab: lane holds nv[rowBase+lo][4k + 2*hi + {0,1}] for k = 0..15
  v2f a[16];
  const float* pa = nv + (rowBase + lo) * C_DIM + 2 * hi;
#pragma unroll
  for (int k = 0; k < 16; k++)
    a[k] = *(const v2f*)(pa + 4 * k);

  int mrow[8];
#pragma unroll
  for (int v = 0; v < 8; v++)
    mrow[v] = meta[rowBase + v + 8 * hi];

  float pos[8], tot[8];
#pragma unroll
  for (int v = 0; v < 8; v++) { pos[v] = 0.0f; tot[v] = 0.0f; }

  for (int tileJ = 0; tileJ < NUM_TILES; tileJ++) {
    const int colBase = tileJ * 16;
    const float* pb = nv + (colBase + lo) * C_DIM + 2 * hi;
    v2f b[16];
#pragma unroll
    for (int k = 0; k < 16; k++)
      b[k] = *(const v2f*)(pb + 4 * k);
    const int cm = meta[colBase + lo];

    v8f acc = {0.f, 0.f, 0.f, 0.f, 0.f, 0.f, 0.f, 0.f};
#pragma unroll
    for (int k = 0; k < 16; k++)
      acc = __builtin_amdgcn_wmma_f32_16x16x4_f32(
          false, a[k], false, b[k], (short)0, acc, false, false);

    const int gcol = colBase + lo;
    const int csel = cm & 4;
    const int clab = cm & 3;
#pragma unroll
    for (int v = 0; v < 8; v++) {
      const int rm = mrow[v];
      const int grow = rowBase + v + 8 * hi;
      // exp(sim/0.07) as single mul + v_exp_f32
      const float e = __builtin_amdgcn_exp2f(acc[v] * EXP2_SCALE);
      const bool pair = (rm & 4) && csel;
      const bool valid = pair && (grow != gcol);
      const float w = valid ? e : 0.0f;
      const bool same = (rm & 3) == clab;
      tot[v] += w;                       // pos + neg combined
      pos[v] += same ? w : 0.0f;
    }
  }

  // Butterfly within each 16-lane half (rows differ between halves).
#pragma unroll
  for (int v = 0; v < 8; v++) {
    float p = pos[v], t = tot[v];
#pragma unroll
    for (int off = 8; off >= 1; off >>= 1) {
      p += __shfl_xor(p, off, 32);
      t += __shfl_xor(t, off, 32);
    }
    if (lo == v) {
      float val = 0.0f;
      if (mrow[v] & 4)
        val = -logf(p / t);
      per[rowBase + v + 8 * hi] = val;
    }
  }
}

// ---------------- Kernel 4: deterministic fixed-order reduction
__global__ __launch_bounds__(1024)
void finalize_kernel(const float* __restrict__ per,
                     const int* __restrict__ nsel,
                     float* __restrict__ out) {
  __shared__ float red[1024];
  const int tid = threadIdx.x;
  float s = 0.0f;
#pragma unroll
  for (int k = 0; k < 8; k++)
    s += per[tid + k * 1024];
  red[tid] = s;
  __syncthreads();
  for (int off = 512; off > 0; off >>= 1) {
    if (tid < off) red[tid] += red[tid + off];
    __syncthreads();
  }
  if (tid == 0) {
    int n = nsel[0];
    if (n < 1) n = 1;
    out[0] = red[0] / (float)n;
  }
}

extern "C" void kernel_launch(void* const* d_in, const int* in_sizes, int n_in,
                              void* d_out, int out_size, void* d_ws, size_t ws_size,
                              hipStream_t stream) {
  (void)in_sizes; (void)n_in; (void)out_size; (void)ws_size;
  const float* feats = (const float*)d_in[0];
  const int* labels  = (const int*)d_in[1];
  float* out = (float*)d_out;

  float* nv  = (float*)d_ws;                       // 8192*64 floats (2 MB)
  int*   meta = (int*)(nv + M_TOTAL * C_DIM);      // 8192 ints
  float* per  = (float*)(meta + M_TOTAL);          // 8192 floats
  int*   nsel = (int*)(per + M_TOTAL);             // 1 int

  setup_meta_kernel<<<dim3(1), dim3(1024), 0, stream>>>(labels, meta, nsel);
  normalize_kernel<<<dim3(M_TOTAL / 256), dim3(256), 0, stream>>>(feats, nv);
  contrast_main_kernel<<<dim3(NUM_TILES / 8), dim3(256), 0, stream>>>(nv, meta, per);
  finalize_kernel<<<dim3(1), dim3(1024), 0, stream>>>(per, nsel, out);
}